// VoxelUnshuffle_82660940579209
// MI455X (gfx1250) — compile-verified
//
#include <hip/hip_runtime.h>
#include <cstdint>

// VoxelUnshuffle (strided pairing): out[n, c*8 + i] = in[n*8 + i, c]
// N voxels, VOLUME=8, C=32 -> per voxel: transpose an 8x32 f32 tile (1KB) in place
// of layout. Pure permutation, HBM-bandwidth bound (~1 GB traffic @ 23.3 TB/s).
//
// Path: global -> LDS via CDNA5 async-LDS loads (ASYNCcnt), conflict-free LDS
// column reads (64 banks), 128-bit coalesced global stores.

#define VOXEL_FLOATS 256        // VOLUME * C
#define VOX_PER_WAVE 4          // 4 KB staged per wave
#define WAVES_PER_BLOCK 8       // 256 threads

__global__ __launch_bounds__(256)
void voxel_unshuffle_kernel(const float* __restrict__ in,
                            float* __restrict__ out,
                            int nvox)
{
    // 8 waves * 4 voxels * 256 floats = 32 KB LDS
    __shared__ __align__(16) float tile[WAVES_PER_BLOCK * VOX_PER_WAVE * VOXEL_FLOATS];

    const int lane  = threadIdx.x & 31;
    const int warp  = threadIdx.x >> 5;
    const int chunk = blockIdx.x * WAVES_PER_BLOCK + warp;
    const int v0    = chunk * VOX_PER_WAVE;
    if (v0 >= nvox) return;

    float* wtile = &tile[warp * (VOX_PER_WAVE * VOXEL_FLOATS)];
    // Generic LDS pointer = {shared_aperture, 32-bit LDS offset}; low 32 bits
    // are the LDS byte address the async-DMA VDST VGPR expects.
    const uint32_t lds_base = (uint32_t)(uintptr_t)wtile;

    if (v0 + VOX_PER_WAVE <= nvox) {
        // ---- Stage 4 KB global -> LDS with async b128 copies ----
        // Each instruction: every lane moves 16 contiguous bytes (wave = 512 B).
        const uint32_t goff = (uint32_t)v0 * 1024u + (uint32_t)lane * 16u;
        const uint32_t loff = lds_base + (uint32_t)lane * 16u;
#pragma unroll
        for (int k = 0; k < 8; ++k) {
            asm volatile(
                "global_load_async_to_lds_b128 %0, %1, %2 offset:0"
                :
                : "v"(loff + (uint32_t)k * 512u),
                  "v"(goff + (uint32_t)k * 512u),
                  "s"(in)
                : "memory");
        }
        asm volatile("s_wait_asynccnt 0x0" ::: "memory");

        // ---- Transpose read + coalesced store ----
        // Lane L gathers column L of each 8x32 tile (stride-32 dwords:
        // for fixed j, lanes hit 32 consecutive dwords -> conflict-free).
#pragma unroll
        for (int q = 0; q < VOX_PER_WAVE; ++q) {
            float r[8];
#pragma unroll
            for (int j = 0; j < 8; ++j)
                r[j] = wtile[q * VOXEL_FLOATS + j * 32 + lane];

            float4* dst = (float4*)(out + (size_t)(v0 + q) * VOXEL_FLOATS + lane * 8);
            dst[0] = make_float4(r[0], r[1], r[2], r[3]);
            dst[1] = make_float4(r[4], r[5], r[6], r[7]);
        }
    } else {
        // Tail (not hit for N=500000, kept for generality): direct coalesced
        // dword gathers, no LDS.
        for (int q = 0; q < VOX_PER_WAVE; ++q) {
            const int v = v0 + q;
            if (v >= nvox) break;
            const float* src = in + (size_t)v * VOXEL_FLOATS;
            float r[8];
#pragma unroll
            for (int j = 0; j < 8; ++j)
                r[j] = src[j * 32 + lane];
            float4* dst = (float4*)(out + (size_t)v * VOXEL_FLOATS + lane * 8);
            dst[0] = make_float4(r[0], r[1], r[2], r[3]);
            dst[1] = make_float4(r[4], r[5], r[6], r[7]);
        }
    }
}

extern "C" void kernel_launch(void* const* d_in, const int* in_sizes, int n_in,
                              void* d_out, int out_size, void* d_ws, size_t ws_size,
                              hipStream_t stream)
{
    (void)n_in; (void)out_size; (void)d_ws; (void)ws_size;

    const float* feat = (const float*)d_in[0];
    const int*   idx  = (const int*)d_in[1];
    float*       out  = (float*)d_out;

    const int total_feat = in_sizes[0];          // N * VOLUME * C
    const int nvox       = total_feat / VOXEL_FLOATS;

    const int chunks = (nvox + VOX_PER_WAVE - 1) / VOX_PER_WAVE;
    const int blocks = (chunks + WAVES_PER_BLOCK - 1) / WAVES_PER_BLOCK;

    voxel_unshuffle_kernel<<<blocks, 256, 0, stream>>>(feat, out, nvox);

    // Second tuple output: indices passed through, appended after the
    // N*256 float main output (raw 4-byte copy; capture-safe async D2D).
    hipMemcpyAsync(out + (size_t)total_feat, idx,
                   (size_t)in_sizes[1] * sizeof(int),
                   hipMemcpyDeviceToDevice, stream);
}